// HighorderNet_57836029608015
// MI455X (gfx1250) — compile-verified
//
#include <hip/hip_runtime.h>

typedef float v2f __attribute__((ext_vector_type(2)));
typedef float v8f __attribute__((ext_vector_type(8)));

#define N_ROWS 1024
#define H_DIM  512
#define O_DIM  512
#define NPAR   4
#define ORDER_C 3
#define KCHUNK 128
#define WAVES_PER_WG 8                 // 256 threads, wave32
#define ROWS_PER_WG  (WAVES_PER_WG * 16)

__global__ __launch_bounds__(256)
void highorder_wmma_kernel(const float* __restrict__ inp,
                           const float* __restrict__ params,
                           const float* __restrict__ bias,
                           const int*   __restrict__ edge_types,
                           const int*   __restrict__ msg_to_p,
                           float*       __restrict__ out)
{
    // B panel staged in LDS, swizzled as [t][k/2][col][k&1] so each lane's
    // (k, k+1) pair for its column is a single 8-byte ds load.
    __shared__ float ldsB[NPAR * KCHUNK * 16];   // 32 KB

    const int tid  = threadIdx.x;
    const int lane = tid & 31;
    const int wave = tid >> 5;
    const int g    = lane >> 4;    // lane half: K offset of 2*g
    const int l16  = lane & 15;

    const int colBase = blockIdx.x * 16;                    // 0..511 step 16
    const int rowBase = blockIdx.y * ROWS_PER_WG + wave*16; // 0..1023 step 16

    const int msg_to = msg_to_p[0];
    // kept factor indices: {0,1,2} \ {msg_to}
    const int i1 = (msg_to == 0) ? 1 : 0;
    const int i2 = (msg_to == 2) ? 1 : 2;
    const float* A1 = inp + (size_t)i1 * N_ROWS * H_DIM;
    const float* A2 = inp + (size_t)i2 * N_ROWS * H_DIM;

    const int myRow = rowBase + l16;
    const float* a1p = A1 + (size_t)myRow * H_DIM;
    const float* a2p = A2 + (size_t)myRow * H_DIM;

    const v8f vzero = {0.f, 0.f, 0.f, 0.f, 0.f, 0.f, 0.f, 0.f};
    v8f acc0 = vzero, acc1 = vzero, acc2 = vzero, acc3 = vzero;

    for (int kc = 0; kc < H_DIM; kc += KCHUNK) {
        // Cooperative load of B chunk: NPAR*KCHUNK*16 floats = 2048 float4's,
        // 8 float4 per thread. Scatter into swizzled LDS layout.
        #pragma unroll
        for (int it = 0; it < 8; ++it) {
            const int fid = tid + it * 256;          // 0..2047
            const int c4  = fid & 3;                 // col group (4 cols)
            const int k   = (fid >> 2) & (KCHUNK-1); // local k
            const int t   = fid >> 9;                // param id
            const float4 v = *(const float4*)(params +
                (((size_t)t * H_DIM + (size_t)(kc + k)) * O_DIM + colBase + c4*4));
            const int base = ((t * (KCHUNK/2) + (k >> 1)) * 16 + c4*4) * 2 + (k & 1);
            ldsB[base + 0] = v.x;
            ldsB[base + 2] = v.y;
            ldsB[base + 4] = v.z;
            ldsB[base + 6] = v.w;
        }
        __syncthreads();

        #pragma unroll 4
        for (int k0 = 0; k0 < KCHUNK; k0 += 4) {
            const int kl = k0 + 2*g;            // local even k this lane owns
            const int kk = kc + kl;             // global k
            v2f a1 = *(const v2f*)(a1p + kk);
            v2f a2 = *(const v2f*)(a2p + kk);
            v2f a  = a1 * a2;                   // fused factor product

            const float* bbase = &ldsB[(kl >> 1) * 32 + l16 * 2];
            v2f b0 = *(const v2f*)(bbase + 0 * (KCHUNK/2) * 32);
            v2f b1 = *(const v2f*)(bbase + 1 * (KCHUNK/2) * 32);
            v2f b2 = *(const v2f*)(bbase + 2 * (KCHUNK/2) * 32);
            v2f b3 = *(const v2f*)(bbase + 3 * (KCHUNK/2) * 32);

            acc0 = __builtin_amdgcn_wmma_f32_16x16x4_f32(false, a, false, b0,
                                                         (short)0, acc0, false, false);
            acc1 = __builtin_amdgcn_wmma_f32_16x16x4_f32(false, a, false, b1,
                                                         (short)0, acc1, false, false);
            acc2 = __builtin_amdgcn_wmma_f32_16x16x4_f32(false, a, false, b2,
                                                         (short)0, acc2, false, false);
            acc3 = __builtin_amdgcn_wmma_f32_16x16x4_f32(false, a, false, b3,
                                                         (short)0, acc3, false, false);
        }
        __syncthreads();
    }

    // Epilogue: per-row select of the right param's accumulator + bias.
    // C/D layout: VGPR j holds row (j + 8*g), col = lane%16.
    const int colGlobal = colBase + l16;
    #pragma unroll
    for (int j = 0; j < 8; ++j) {
        const int row = rowBase + j + 8*g;
        const int id  = edge_types[row * ORDER_C + msg_to];
        float v = acc0[j];
        v = (id == 1) ? acc1[j] : v;
        v = (id == 2) ? acc2[j] : v;
        v = (id == 3) ? acc3[j] : v;
        v += bias[id * O_DIM + colGlobal];
        out[(size_t)row * O_DIM + colGlobal] = v;
    }
}

extern "C" void kernel_launch(void* const* d_in, const int* in_sizes, int n_in,
                              void* d_out, int out_size, void* d_ws, size_t ws_size,
                              hipStream_t stream) {
    const float* inp        = (const float*)d_in[0];
    const float* params     = (const float*)d_in[1];
    const float* bias       = (const float*)d_in[2];
    const int*   edge_types = (const int*)d_in[3];
    const int*   msg_to     = (const int*)d_in[6];
    float* out = (float*)d_out;

    dim3 grid(O_DIM / 16, N_ROWS / ROWS_PER_WG);  // (32, 8) workgroups
    dim3 block(256);                               // 8 wave32 per WG
    highorder_wmma_kernel<<<grid, block, 0, stream>>>(inp, params, bias,
                                                      edge_types, msg_to, out);
}